// CausalBoW_24670292149167
// MI455X (gfx1250) — compile-verified
//
#include <hip/hip_runtime.h>

// Causal BoW (running cumulative mean) via a pure-WMMA tile-prefix scan.
//
// out[b,t,c] = (1/(t+1)) * sum_{s<=t} x[b,s,c]
//
// Per 16x16 (TxC) tile X:
//   D = L @ X + S_prev     L = 16x16 lower-triangular ones (intra-tile prefix)
//   S = 1 @ X + S_prev     1 = 16x16 all-ones  (running column sums)
// Every row of S equals the running column sum, so every element of S is the
// per-column carry in BOTH lane halves -> S feeds straight into the next
// tile's C operand. No cross-lane ops, no LDS; carry stays in the WMMA domain.
// All via chained V_WMMA_F32_16X16X4_F32 at full fp32 precision.

typedef float v2f __attribute__((ext_vector_type(2)));
typedef float v8f __attribute__((ext_vector_type(8)));

__global__ __launch_bounds__(256)
void causal_bow_wmma_kernel(const float* __restrict__ x,
                            float* __restrict__ out) {
  constexpr int T = 2048;
  constexpr int C = 512;

  const unsigned lane = threadIdx.x & 31u;
  const unsigned waveInBlock = threadIdx.x >> 5;
  const unsigned wave = blockIdx.x * (blockDim.x >> 5) + waveInBlock;
  constexpr unsigned nC16 = (unsigned)C >> 4;  // 32 column groups
  const unsigned b    = wave / nC16;           // batch index
  const unsigned c16  = wave % nC16;           // column-group index

  const unsigned col    = lane & 15u;          // N (and M for A-matrix)
  const bool     hiHalf = lane >= 16u;
  const unsigned hi2    = hiHalf ? 2u : 0u;    // A/B K-row offset by lane half
  const unsigned hi8    = hiHalf ? 8u : 0u;    // C/D M-row offset by lane half

  // L = lower-triangular ones, as four 16x4 f32 chunks.
  // A layout (32-bit 16x4): lanes 0-15 / 16-31 both hold M=0..15;
  // VGPR0 -> K = {0 | 2}, VGPR1 -> K = {1 | 3} by lane half.
  v2f a[4];
#pragma unroll
  for (int j = 0; j < 4; ++j) {
    unsigned k0 = 4u * (unsigned)j + hi2;
    a[j].x = (k0 <= col)      ? 1.0f : 0.0f;
    a[j].y = (k0 + 1u <= col) ? 1.0f : 0.0f;
  }
  const v2f onesA = {1.0f, 1.0f};              // all-ones A chunk (any K slice)

  // Fold lane-half row offsets into base pointers so every unrolled access
  // uses a compile-time immediate byte offset (single base VGPR per tile).
  const size_t waveBase = (size_t)b * (size_t)T * (size_t)C + c16 * 16u + col;
  const float* xp = x   + waveBase + (size_t)hi2 * C;  // load base (B rows)
  float*       op = out + waveBase + (size_t)hi8 * C;  // store base (D rows)

  // Denominators t+1 for rows M = hi8+i; exact in f32 (max 2048), +16/tile.
  float denom[8];
#pragma unroll
  for (int i = 0; i < 8; ++i) denom[i] = (float)(hi8 + (unsigned)i + 1u);

  v8f s = {};                                  // running column sums (carry)
  constexpr int nTiles = T >> 4;               // 128 tiles of 16 timesteps

  for (int tile = 0; tile < nTiles; ++tile) {
    // Load B tile (16x16 f32): chunk j covers K rows 4j..4j+3.
    // VGPR0 -> K = {4j | 4j+2}, VGPR1 -> K = {4j+1 | 4j+3} by lane half
    // (hi2 folded into xp). All offsets are immediates -> one s_clause.
    v2f bb[4];
#pragma unroll
    for (int j = 0; j < 4; ++j) {
      bb[j].x = xp[(4 * j) * C];
      bb[j].y = xp[(4 * j + 1) * C];
    }

    // Prefetch next tile's stream (global_prefetch_b8).
    if (tile + 1 < nTiles)
      __builtin_prefetch(xp + 16 * C, 0, 3);

    // D = L @ X + S_prev : carry enters through the C operand of the first
    // WMMA, then chain K over the tile.
    v8f d = __builtin_amdgcn_wmma_f32_16x16x4_f32(
        false, a[0], false, bb[0], (short)0, s, false, false);
#pragma unroll
    for (int j = 1; j < 4; ++j) {
      d = __builtin_amdgcn_wmma_f32_16x16x4_f32(
          false, a[j], false, bb[j], (short)0, d, false, false);
    }

    // S += 1 @ X : update running column sums (carry for the next tile).
#pragma unroll
    for (int j = 0; j < 4; ++j) {
      s = __builtin_amdgcn_wmma_f32_16x16x4_f32(
          false, onesA, false, bb[j], (short)0, s, false, false);
    }

    // Scale row M=hi8+i by 1/denom and store (immediate offsets; hi8 folded
    // into op). Reciprocal: v_rcp_f32 + one NR refinement (2 FMAs).
#pragma unroll
    for (int i = 0; i < 8; ++i) {
      float dn  = denom[i];
      float r   = __builtin_amdgcn_rcpf(dn);
      r = __builtin_fmaf(__builtin_fmaf(-dn, r, 1.0f), r, r);
      op[i * C] = d[i] * r;
      denom[i]  = dn + 16.0f;
    }

    xp += 16 * C;
    op += 16 * C;
  }
}

extern "C" void kernel_launch(void* const* d_in, const int* in_sizes, int n_in,
                              void* d_out, int out_size, void* d_ws, size_t ws_size,
                              hipStream_t stream) {
  (void)in_sizes; (void)n_in; (void)d_ws; (void)ws_size; (void)out_size;

  const float* x = (const float*)d_in[0];
  float* out = (float*)d_out;

  const int B = 32, C = 512;
  const int nC16 = C / 16;                 // 32
  const int waves = B * nC16;              // 1024 waves
  const int threads = waves * 32;          // 32768 threads
  const int block = 256;                   // 8 waves/block
  const int grid = threads / block;        // 128 blocks

  causal_bow_wmma_kernel<<<grid, block, 0, stream>>>(x, out);
}